// EnhancedCoordAtt_78855599555233
// MI455X (gfx1250) — compile-verified
//
#include <hip/hip_runtime.h>
#include <hip/hip_bf16.h>
#include <math.h>

// ---------------------------------------------------------------------------
// EnhancedCoordAtt for MI455X (gfx1250, wave32).
// Memory-bound problem (~420 MB total traffic -> ~18us floor @ 23.3 TB/s).
// fp32 throughout; WMMA (V_WMMA_F32_16X16X4_F32) used for the 8->256
// channel-mix einsums which are genuine GEMMs (M=256, K=8, N=2048 per side).
// Sigmoids use the hardware v_rcp_f32 (approximate reciprocal) instead of
// the IEEE div sequence -- these are gate values, not accumulators.
// ---------------------------------------------------------------------------

typedef __attribute__((ext_vector_type(2))) float v2f;
typedef __attribute__((ext_vector_type(8))) float v8f;

#define N_BATCH 32
#define C_IN    256
#define HW      64          // h == w == 64
#define MIP     8
#define OUP     256
#define CI3     768         // 3 * C_IN
#define ROWS    128         // h + w
#define BN_EPS  1e-5f

__device__ __forceinline__ float sigmoidf_(float v) {
    // v_exp_f32 + v_add_f32 + v_rcp_f32 (no div_scale/div_fixup chain)
    return __builtin_amdgcn_rcpf(1.0f + __expf(-v));
}

// ---------------------------------------------------------------------------
// Kernel 1: directional pooling.  One block per (n,c).  64x64 tile in LDS
// with stride-65 padding => conflict-free row reads AND column reads.
// Output yin layout: [n][row 0..127][ci 0..767], ci blocks:
//   row<64  : {row-mean-full, row-mean-left, row-mean-right} for height stats
//   row>=64 : {col-mean-full, col-mean-top,  col-mean-bottom} (xw part)
// ---------------------------------------------------------------------------
__global__ __launch_bounds__(256)
void pool_kernel(const float* __restrict__ x, float* __restrict__ yin) {
    __shared__ float tile[HW * 65];

    const int n = blockIdx.x >> 8;          // /256
    const int c = blockIdx.x & 255;
    const int t = threadIdx.x;

    const float4* src = (const float4*)(x + ((size_t)(n * C_IN + c)) * (HW * HW));
    // load 64x64 tile: 1024 float4, 256 threads x 4
    #pragma unroll
    for (int k = 0; k < 4; ++k) {
        int f4 = k * 256 + t;
        float4 v = src[f4];
        int e   = f4 * 4;
        int row = e >> 6;
        int col = e & 63;
        float* p = &tile[row * 65 + col];
        p[0] = v.x; p[1] = v.y; p[2] = v.z; p[3] = v.w;
    }
    __syncthreads();

    if (t < 64) {
        // row statistics for row r = t
        const int r = t;
        float sf = 0.f, s1 = 0.f, s2 = 0.f;
        #pragma unroll 8
        for (int i = 0; i < 64; ++i) {
            float v = tile[r * 65 + i];
            sf += v;
            if (i < 32) s1 += v; else s2 += v;
        }
        float* dst = yin + ((size_t)n * ROWS + r) * CI3;
        dst[c]            = sf * (1.0f / 64.0f);
        dst[C_IN + c]     = s1 * (1.0f / 32.0f);
        dst[2 * C_IN + c] = s2 * (1.0f / 32.0f);
    } else if (t < 128) {
        // column statistics for column cc = t-64  -> stored at row 64+cc
        const int cc = t - 64;
        float sf = 0.f, s1 = 0.f, s2 = 0.f;
        #pragma unroll 8
        for (int i = 0; i < 64; ++i) {
            float v = tile[i * 65 + cc];
            sf += v;
            if (i < 32) s1 += v; else s2 += v;
        }
        float* dst = yin + ((size_t)n * ROWS + 64 + cc) * CI3;
        dst[c]            = sf * (1.0f / 64.0f);
        dst[C_IN + c]     = s1 * (1.0f / 32.0f);
        dst[2 * C_IN + c] = s2 * (1.0f / 32.0f);
    }
}

// ---------------------------------------------------------------------------
// Kernel 2: degenerate dilated conv (3 height taps, kw=1 only) + bias + BN +
// SiLU.  One block per (n, r): 4096 blocks x 64 threads.  Stages the 3
// needed yin rows in LDS (coalesced), splits K=768 across lanes, LDS reduce.
// Output yconv: [n][o 0..7][r 0..127] (post-SiLU, pre-SE).
// ---------------------------------------------------------------------------
__global__ __launch_bounds__(64)
void conv_bn_silu_kernel(const float* __restrict__ yin,
                         const float* __restrict__ w,     // (8,768,3,3)
                         const float* __restrict__ bias,
                         const float* __restrict__ gamma,
                         const float* __restrict__ beta,
                         const float* __restrict__ mean,
                         const float* __restrict__ var,
                         float* __restrict__ yconv) {
    __shared__ float lrow[3 * CI3];     // 9 KB
    __shared__ float red[MIP * 64];

    const int n = blockIdx.x >> 7;
    const int r = blockIdx.x & 127;
    const int t = threadIdx.x;

    // stage rows r-2, r, r+2 (zero-padded) -- coalesced loads
    #pragma unroll
    for (int kh = 0; kh < 3; ++kh) {
        int rr = r - 2 + 2 * kh;
        const bool ok = (rr >= 0) && (rr < ROWS);
        const float* srow = yin + ((size_t)n * ROWS + (ok ? rr : 0)) * CI3;
        for (int ci = t; ci < CI3; ci += 64)
            lrow[kh * CI3 + ci] = ok ? srow[ci] : 0.0f;
    }
    __syncthreads();

    float acc[MIP];
    #pragma unroll
    for (int o = 0; o < MIP; ++o) acc[o] = 0.0f;

    #pragma unroll
    for (int kh = 0; kh < 3; ++kh) {
        for (int ci = t; ci < CI3; ci += 64) {
            float yv = lrow[kh * CI3 + ci];
            #pragma unroll
            for (int o = 0; o < MIP; ++o) {
                // w[o][ci][kh][1]
                acc[o] = fmaf(yv, w[((size_t)(o * CI3 + ci)) * 9 + kh * 3 + 1], acc[o]);
            }
        }
    }
    #pragma unroll
    for (int o = 0; o < MIP; ++o) red[o * 64 + t] = acc[o];
    __syncthreads();

    if (t < MIP) {
        const int o = t;
        float s = 0.0f;
        #pragma unroll 8
        for (int i = 0; i < 64; ++i) s += red[o * 64 + i];
        s += bias[o];
        float sc = gamma[o] * rsqrtf(var[o] + BN_EPS);
        s = (s - mean[o]) * sc + beta[o];
        s = s * sigmoidf_(s);                       // SiLU
        yconv[((size_t)n * MIP + o) * ROWS + r] = s;
    }
}

// ---------------------------------------------------------------------------
// Kernel 3: SE gate.  One block per n (128 threads).  mean over the 128 rows
// per (n,o), sigmoid gate, scale -> yact[n][8][128].
// ---------------------------------------------------------------------------
__global__ __launch_bounds__(128)
void se_scale_kernel(const float* __restrict__ yconv,
                     const float* __restrict__ gate_w,
                     const float* __restrict__ gate_b,
                     float* __restrict__ yact) {
    __shared__ float red[128];
    __shared__ float se[MIP];

    const int n = blockIdx.x;
    const int t = threadIdx.x;
    const int o = t >> 4;           // 16 threads per output channel
    const int ri = t & 15;

    const float* base = yconv + (size_t)n * MIP * ROWS;
    float p = 0.0f;
    #pragma unroll
    for (int k = 0; k < 8; ++k) p += base[o * ROWS + ri + 16 * k];
    red[t] = p;
    __syncthreads();

    if (ri == 0) {
        float s = 0.0f;
        #pragma unroll
        for (int i = 0; i < 16; ++i) s += red[o * 16 + i];
        se[o] = sigmoidf_(gate_w[0] * (s * (1.0f / 128.0f)) + gate_b[0]);
    }
    __syncthreads();

    float* dst = yact + (size_t)n * MIP * ROWS;
    for (int u = t; u < MIP * ROWS; u += 128) {
        int oo = u >> 7;            // u / 128
        dst[u] = base[u] * se[oo];
    }
}

// ---------------------------------------------------------------------------
// Kernel 4: channel-mix einsums as WMMA GEMMs + fused bias + sigmoid.
//   side 0: ah[n][o][r] = sigmoid( sum_c wh[o][c]*yact[n][c][r]      + bh[o] )
//   side 1: aw[n][o][r] = sigmoid( sum_c ww[o][c]*yact[n][c][64+r]   + bw[o] )
// M=256 (o), K=8, N=2048 (n*64).  One 16x16 output tile per wave via two
// chained V_WMMA_F32_16X16X4_F32 (K=4 each).  4096 tiles total.
// A layout (16x4 f32): lane = M (0..15 mirrored), vgpr pair = K half.
// C/D layout: col = lane&15, row = vgpr + 8*(lane>>4).
// ---------------------------------------------------------------------------
__global__ __launch_bounds__(256)
void einsum_wmma_kernel(const float* __restrict__ yact,   // [32][8][128]
                        const float* __restrict__ wh, const float* __restrict__ bh,
                        const float* __restrict__ ww, const float* __restrict__ bw,
                        float* __restrict__ ah, float* __restrict__ aw) {
    const int wave = threadIdx.x >> 5;
    const int lane = threadIdx.x & 31;
    const int tilei = blockIdx.x * 8 + wave;   // 0..4095
    const int side = tilei >> 11;              // 0 = h side, 1 = w side
    const int rem  = tilei & 2047;
    const int o0   = (rem >> 7) * 16;          // output-channel tile base
    const int j0   = (rem & 127) * 16;         // column tile base (j = n*64+r)

    const int m  = lane & 15;
    const int hi = lane >> 4;                  // K-half selector (A/B), row-half (D)

    const float* W = side ? ww : wh;           // (256, 8)
    const float* B = side ? bw : bh;

    // A: 16x4 fragments, rows o0+m, K halves {0,1} and {2,3} of K=8
    const float* wrow = W + (size_t)(o0 + m) * MIP;
    v2f a0, a1;
    a0.x = wrow[2 * hi + 0];
    a0.y = wrow[2 * hi + 1];
    a1.x = wrow[4 + 2 * hi + 0];
    a1.y = wrow[4 + 2 * hi + 1];

    // B: 4x16 fragments, column j0+m
    const int j = j0 + m;
    const int n = j >> 6;
    const int r = (j & 63) + (side ? HW : 0);
    const float* ybase = yact + (size_t)n * MIP * ROWS + r;   // + c*ROWS
    v2f b0, b1;
    b0.x = ybase[(2 * hi + 0) * ROWS];
    b0.y = ybase[(2 * hi + 1) * ROWS];
    b1.x = ybase[(4 + 2 * hi) * ROWS];
    b1.y = ybase[(5 + 2 * hi) * ROWS];

    v8f c = {};
    c = __builtin_amdgcn_wmma_f32_16x16x4_f32(false, a0, false, b0, (short)0, c, false, false);
    c = __builtin_amdgcn_wmma_f32_16x16x4_f32(false, a1, false, b1, (short)0, c, false, false);

    float* OUT = side ? aw : ah;               // [32][256][64]
    const int rr = j & 63;
    #pragma unroll
    for (int v = 0; v < 8; ++v) {
        int orow = o0 + v + 8 * hi;
        float val = sigmoidf_(c[v] + B[orow]);
        OUT[((size_t)n * OUP + orow) * HW + rr] = val;
    }
}

// ---------------------------------------------------------------------------
// Kernel 5: out = x * ah * aw   (float4, one f4 per thread; 8.4M threads)
// ---------------------------------------------------------------------------
__global__ __launch_bounds__(256)
void final_kernel(const float* __restrict__ x,
                  const float* __restrict__ ah,
                  const float* __restrict__ aw,
                  float* __restrict__ out) {
    const int idx4 = blockIdx.x * 256 + threadIdx.x;   // < 8,388,608
    const int j4 = idx4 & 15;
    const int i  = (idx4 >> 4) & 63;
    const int ch = (idx4 >> 10) & 255;
    const int n  = idx4 >> 18;

    float4 xv = ((const float4*)x)[idx4];
    float  hs = ah[((size_t)n * OUP + ch) * HW + i];
    float4 wv = ((const float4*)aw)[((size_t)n * OUP + ch) * 16 + j4];

    float4 o;
    o.x = xv.x * hs * wv.x;
    o.y = xv.y * hs * wv.y;
    o.z = xv.z * hs * wv.z;
    o.w = xv.w * hs * wv.w;
    ((float4*)out)[idx4] = o;
}

// ---------------------------------------------------------------------------
extern "C" void kernel_launch(void* const* d_in, const int* in_sizes, int n_in,
                              void* d_out, int out_size, void* d_ws, size_t ws_size,
                              hipStream_t stream) {
    const float* x       = (const float*)d_in[0];
    const float* conv1_w = (const float*)d_in[1];
    const float* conv1_b = (const float*)d_in[2];
    const float* bn_g    = (const float*)d_in[3];
    const float* bn_b    = (const float*)d_in[4];
    const float* bn_m    = (const float*)d_in[5];
    const float* bn_v    = (const float*)d_in[6];
    const float* gate_w  = (const float*)d_in[7];
    const float* gate_b  = (const float*)d_in[8];
    const float* convh_w = (const float*)d_in[9];
    const float* convh_b = (const float*)d_in[10];
    const float* convw_w = (const float*)d_in[11];
    const float* convw_b = (const float*)d_in[12];
    float* out = (float*)d_out;

    // workspace carve-out (~16.3 MB)
    char* ws = (char*)d_ws;
    float* yin   = (float*)(ws);                                   // 32*128*768 = 12.0 MB
    float* yconv = (float*)(ws + 12582912);                        // 32*8*128   = 128 KB
    float* yact  = (float*)(ws + 12582912 + 131072);               // 128 KB
    float* ah    = (float*)(ws + 12582912 + 2 * 131072);           // 32*256*64  = 2 MB
    float* aw    = (float*)(ws + 12582912 + 2 * 131072 + 2097152); // 2 MB

    pool_kernel<<<N_BATCH * C_IN, 256, 0, stream>>>(x, yin);

    conv_bn_silu_kernel<<<N_BATCH * ROWS, 64, 0, stream>>>(
        yin, conv1_w, conv1_b, bn_g, bn_b, bn_m, bn_v, yconv);

    se_scale_kernel<<<N_BATCH, 128, 0, stream>>>(yconv, gate_w, gate_b, yact);

    einsum_wmma_kernel<<<512, 256, 0, stream>>>(
        yact, convh_w, convh_b, convw_w, convw_b, ah, aw);

    final_kernel<<<(N_BATCH * C_IN * HW * HW / 4) / 256, 256, 0, stream>>>(
        x, ah, aw, out);
}